// EmbeddingGATEdge_24575802867853
// MI455X (gfx1250) — compile-verified
//
#include <hip/hip_runtime.h>
#include <hip/hip_bf16.h>

#define N_NODES   25000
#define N_EDGES   400000
#define N_NDICT   100
#define N_EDICT   16
#define EMB       128
#define HEADS     4
#define F_OUT     32
#define HF        128
#define NDICT_PAD 112          // 100 rounded up to 16 for WMMA K
#define NEG_SLOPE 0.2f
#define LN_EPS    1e-5f

typedef float v2f __attribute__((ext_vector_type(2)));
typedef float v8f __attribute__((ext_vector_type(8)));

// ---- workspace layout (bytes) ----
static constexpr size_t OFF_PROJ_NODE = 0;                      // 112*128 f32
static constexpr size_t OFF_PROJ_EDGE = 57344;                  // 16*128 f32
static constexpr size_t OFF_LN_EDGE   = 65536;                  // 16*128 f32
static constexpr size_t OFF_ATT1      = 73728;                  // 100*4 f32
static constexpr size_t OFF_ATT2      = 75776;                  // 100*4 f32
static constexpr size_t OFF_ATT3      = 77824;                  // 16*4 f32
static constexpr size_t OFF_SEGMAX    = 78336;                  // 25000*4 u32 (ordered-float)
static constexpr size_t OFF_SEGSUM    = 478336;                 // 25000*4 f32
static constexpr size_t OFF_INDEG     = 878336;                 // 25000 i32
static constexpr size_t OFF_EIJ       = 978336;                 // 400000*4 f32
static constexpr size_t OFF_WEIGHT    = 7378432;                // 4*25000*112 f32 (8B aligned)
static constexpr size_t WS_FACTORED   = OFF_WEIGHT + (size_t)HEADS * N_NODES * NDICT_PAD * 4;

// ordered-float mapping: monotone f32 -> u32 so atomicMax works for any sign
__device__ __forceinline__ unsigned f2ord(float f) {
    unsigned u = __float_as_uint(f);
    return (u & 0x80000000u) ? ~u : (u | 0x80000000u);
}
__device__ __forceinline__ float ord2f(unsigned u) {
    return __uint_as_float((u & 0x80000000u) ? (u & 0x7FFFFFFFu) : ~u);
}

// ---------------------------------------------------------------------------
// K1: dictionary GEMM via V_WMMA_F32_16X16X4_F32.
// C[m,n] = sum_k A[m,k]*B[k,n] + bias[n]; A is arows x 128; C stored
// crows x 128 (crows is a multiple of 16 == 16*gridDim.y; padding rows
// written as 0 via branch-free cndmask).  One wave per 16x16 tile.
// A row addresses are CLAMPED (never predicated) -> no exec-mask branches.
// ---------------------------------------------------------------------------
__global__ void k_wmma_dict_gemm(const float* __restrict__ A, int arows,
                                 const float* __restrict__ B,
                                 const float* __restrict__ bias,
                                 float* __restrict__ C) {
    const int lane = threadIdx.x;           // 0..31
    const int half = lane >> 4;             // K-pair select
    const int lm   = lane & 15;
    const int m0   = blockIdx.y * 16;
    const int n0   = blockIdx.x * 16;
    const int arow = min(m0 + lm, arows - 1);   // clamped, always in-bounds
    const int ncol = n0 + lm;
    v8f acc = {};
    #pragma unroll 4
    for (int k0 = 0; k0 < EMB; k0 += 4) {
        const int ka = k0 + 2 * half;           // even -> 8B-aligned pair
        const float2 av = *(const float2*)(A + arow * EMB + ka);
        v2f a, b;
        a.x = av.x; a.y = av.y;
        b.x = B[ka * HF + ncol];
        b.y = B[(ka + 1) * HF + ncol];
        acc = __builtin_amdgcn_wmma_f32_16x16x4_f32(false, a, false, b,
                                                    (short)0, acc, false, false);
    }
    const float bv = bias[ncol];
    #pragma unroll
    for (int r = 0; r < 8; ++r) {
        const int row = m0 + r + 8 * half;      // always < crows by grid
        C[row * HF + ncol] = (row < arows) ? (acc[r] + bv) : 0.0f;  // cndmask
    }
}

// ---------------------------------------------------------------------------
// K2a: per node-dict row, per-head dot with a1 (tgt) and a2 (src)
// ---------------------------------------------------------------------------
__global__ void k_att_node(const float* __restrict__ proj_node,
                           const float* __restrict__ a_w,
                           float* __restrict__ att1, float* __restrict__ att2) {
    const int d = blockIdx.x;               // dict row
    const int t = threadIdx.x;              // 0..127
    const int lane = t & 31;                // feature within head
    const int head = t >> 5;                // one wave per head
    const float v = proj_node[d * HF + t];
    float s1 = v * a_w[lane];
    float s2 = v * a_w[F_OUT + lane];
    for (int o = 16; o; o >>= 1) { s1 += __shfl_xor(s1, o, 32); s2 += __shfl_xor(s2, o, 32); }
    if (lane == 0) { att1[d * HEADS + head] = s1; att2[d * HEADS + head] = s2; }
}

// ---------------------------------------------------------------------------
// K2b: per edge-dict row: per-head dot with a3, plus LayerNorm table
// ---------------------------------------------------------------------------
__device__ __forceinline__ float blockSum128(float v, float* sbuf) {
    for (int o = 16; o; o >>= 1) v += __shfl_xor(v, o, 32);
    const int w = threadIdx.x >> 5;
    if ((threadIdx.x & 31) == 0) sbuf[w] = v;
    __syncthreads();
    const float tot = sbuf[0] + sbuf[1] + sbuf[2] + sbuf[3];
    __syncthreads();
    return tot;
}

__global__ void k_edge_tables(const float* __restrict__ proj_edge,
                              const float* __restrict__ a_w,
                              const float* __restrict__ gamma,
                              const float* __restrict__ beta,
                              float* __restrict__ att3,
                              float* __restrict__ ln_edge) {
    __shared__ float sbuf[4];
    const int d = blockIdx.x;
    const int t = threadIdx.x;
    const int lane = t & 31;
    const int head = t >> 5;
    const float v = proj_edge[d * HF + t];
    float s3 = v * a_w[2 * F_OUT + lane];
    for (int o = 16; o; o >>= 1) s3 += __shfl_xor(s3, o, 32);
    if (lane == 0) att3[d * HEADS + head] = s3;
    const float mu  = blockSum128(v, sbuf) * (1.0f / HF);
    const float dv  = v - mu;
    const float var = blockSum128(dv * dv, sbuf) * (1.0f / HF);
    ln_edge[d * HF + t] = dv * rsqrtf(var + LN_EPS) * gamma[t] + beta[t];
}

// ---------------------------------------------------------------------------
// K3: per-edge logits + leaky relu + segment max (ordered-uint atomicMax),
//     and in-degree count
// ---------------------------------------------------------------------------
__global__ void k_logits(const int* __restrict__ edge_index,
                         const int* __restrict__ edge_feat,
                         const int* __restrict__ node_feat,
                         const float* __restrict__ att1,
                         const float* __restrict__ att2,
                         const float* __restrict__ att3,
                         const float* __restrict__ a_b,
                         float* __restrict__ eij,
                         unsigned* __restrict__ segmax,
                         int* __restrict__ indeg) {
    const int e = blockIdx.x * blockDim.x + threadIdx.x;
    if (e >= N_EDGES) return;
    const int src = edge_index[e];
    const int tgt = edge_index[N_EDGES + e];
    const int ef  = edge_feat[e];
    const int ns  = node_feat[src];
    const int nt  = node_feat[tgt];
    atomicAdd(&indeg[tgt], 1);
    const float ab = a_b[0];
    #pragma unroll
    for (int h = 0; h < HEADS; ++h) {
        float l = att1[nt * HEADS + h] + att2[ns * HEADS + h] + att3[ef * HEADS + h] + ab;
        l = (l >= 0.0f) ? l : NEG_SLOPE * l;
        eij[e * HEADS + h] = l;
        atomicMax(&segmax[tgt * HEADS + h], f2ord(l));
    }
}

// ---------------------------------------------------------------------------
// K3b: stream edge_out = ln_edge[edge_feat[e]]  (float4 vectorized gather)
// ---------------------------------------------------------------------------
__global__ void k_edge_out(const int* __restrict__ edge_feat,
                           const float4* __restrict__ ln_edge4,
                           float4* __restrict__ edge_out4) {
    const int idx = blockIdx.x * blockDim.x + threadIdx.x;   // NE * 32
    if (idx >= N_EDGES * (HF / 4)) return;
    const int e = idx >> 5;
    const int q = idx & 31;
    edge_out4[e * 32 + q] = ln_edge4[edge_feat[e] * 32 + q];
}

// ---------------------------------------------------------------------------
// K4: ex = exp(eij - segmax[tgt]) (in place) ; segment sum via atomicAdd
// ---------------------------------------------------------------------------
__global__ void k_expsum(const int* __restrict__ edge_index,
                         float* __restrict__ eij,
                         const unsigned* __restrict__ segmax,
                         float* __restrict__ segsum) {
    const int idx = blockIdx.x * blockDim.x + threadIdx.x;   // NE * HEADS
    if (idx >= N_EDGES * HEADS) return;
    const int e = idx >> 2;
    const int h = idx & 3;
    const int tgt = edge_index[N_EDGES + e];
    const float m = ord2f(segmax[tgt * HEADS + h]);
    const float ex = expf(eij[idx] - m);
    eij[idx] = ex;
    atomicAdd(&segsum[tgt * HEADS + h], ex);
}

// ---------------------------------------------------------------------------
// K5 (factored): accumulate alpha into per-(head,target,src-dict) weights
// ---------------------------------------------------------------------------
__global__ void k_alpha_weights(const int* __restrict__ edge_index,
                                const int* __restrict__ node_feat,
                                const float* __restrict__ eij,
                                const float* __restrict__ segsum,
                                float* __restrict__ weight) {
    const int idx = blockIdx.x * blockDim.x + threadIdx.x;   // NE * HEADS
    if (idx >= N_EDGES * HEADS) return;
    const int e = idx >> 2;
    const int h = idx & 3;
    const int tgt = edge_index[N_EDGES + e];
    const float alpha = eij[idx] / segsum[tgt * HEADS + h];
    const int ds = node_feat[edge_index[e]];                 // src dict id
    atomicAdd(&weight[((size_t)h * N_NODES + tgt) * NDICT_PAD + ds], alpha);
}

// ---------------------------------------------------------------------------
// K6 (factored): out_h(25000x32) = W_h(25000x112) @ P_h(112x32), WMMA f32.
// grid = (HEADS, 1563).  Each wave computes BOTH 16-col halves of one head
// tile row: one A fragment feeds two WMMAs (halves traffic on the 45MB
// weight matrix, doubles matrix-op density).  A row addresses clamped.
// Store epilogue: wave-uniform fast path (full tiles, unconditional stores);
// only the single partial last tile takes the guarded path.
// ---------------------------------------------------------------------------
__global__ void k_wmma_recover(const float* __restrict__ weight,
                               const float* __restrict__ proj_node,
                               float* __restrict__ out_node) {
    const int lane = threadIdx.x;
    const int half = lane >> 4;
    const int lm   = lane & 15;
    const int head = blockIdx.x;
    const int m0   = blockIdx.y * 16;
    const int arow = min(m0 + lm, N_NODES - 1);       // clamped A row
    const int colL = head * F_OUT + lm;               // cols [h*32, h*32+16)
    const int colH = colL + 16;                       // cols [h*32+16, h*32+32)
    const float* Wh = weight + (size_t)head * N_NODES * NDICT_PAD;
    v8f accL = {}, accH = {};
    #pragma unroll 4
    for (int k0 = 0; k0 < NDICT_PAD; k0 += 4) {
        const int ka = k0 + 2 * half;                 // even -> 8B-aligned pair
        const float2 av = *(const float2*)(Wh + (size_t)arow * NDICT_PAD + ka);
        v2f a, bL, bH;
        a.x = av.x; a.y = av.y;
        bL.x = proj_node[ka * HF + colL];
        bL.y = proj_node[(ka + 1) * HF + colL];
        bH.x = proj_node[ka * HF + colH];
        bH.y = proj_node[(ka + 1) * HF + colH];
        accL = __builtin_amdgcn_wmma_f32_16x16x4_f32(false, a, false, bL,
                                                     (short)0, accL, false, false);
        accH = __builtin_amdgcn_wmma_f32_16x16x4_f32(false, a, false, bH,
                                                     (short)0, accH, false, false);
    }
    if (m0 + 16 <= N_NODES) {                         // uniform fast path
        #pragma unroll
        for (int r = 0; r < 8; ++r) {
            const int row = m0 + r + 8 * half;
            out_node[row * HF + colL] = accL[r];
            out_node[row * HF + colH] = accH[r];
        }
    } else {                                          // one partial tile
        #pragma unroll
        for (int r = 0; r < 8; ++r) {
            const int row = m0 + r + 8 * half;
            if (row < N_NODES) {
                out_node[row * HF + colL] = accL[r];
                out_node[row * HF + colH] = accH[r];
            }
        }
    }
}

// ---------------------------------------------------------------------------
// K5' (fallback): direct message scatter  out[tgt] += alpha_h * h_j
// ---------------------------------------------------------------------------
__global__ void k_msg_direct(const int* __restrict__ edge_index,
                             const int* __restrict__ node_feat,
                             const float* __restrict__ eij,
                             const float* __restrict__ segsum,
                             const float* __restrict__ proj_node,
                             float* __restrict__ out_node) {
    const int idx = blockIdx.x * blockDim.x + threadIdx.x;   // NE * 128
    if (idx >= N_EDGES * HF) return;
    const int e = idx >> 7;
    const int f = idx & 127;
    const int h = f >> 5;
    const int tgt = edge_index[N_EDGES + e];
    const float alpha = eij[e * HEADS + h] / segsum[tgt * HEADS + h];
    const int ds = node_feat[edge_index[e]];
    atomicAdd(&out_node[tgt * HF + f], alpha * proj_node[ds * HF + f]);
}

// ---------------------------------------------------------------------------
// K7: add indeg(t) * h_i(t), then LayerNorm in place. One block per node row.
// ---------------------------------------------------------------------------
__global__ void k_finalize_node(const int* __restrict__ node_feat,
                                const int* __restrict__ indeg,
                                const float* __restrict__ proj_node,
                                const float* __restrict__ gamma,
                                const float* __restrict__ beta,
                                float* __restrict__ out_node) {
    __shared__ float sbuf[4];
    const int t = blockIdx.x;
    const int f = threadIdx.x;
    float v = out_node[t * HF + f]
            + (float)indeg[t] * proj_node[node_feat[t] * HF + f];
    const float mu  = blockSum128(v, sbuf) * (1.0f / HF);
    const float dv  = v - mu;
    const float var = blockSum128(dv * dv, sbuf) * (1.0f / HF);
    out_node[t * HF + f] = dv * rsqrtf(var + LN_EPS) * gamma[f] + beta[f];
}

// ---------------------------------------------------------------------------
extern "C" void kernel_launch(void* const* d_in, const int* in_sizes, int n_in,
                              void* d_out, int out_size, void* d_ws, size_t ws_size,
                              hipStream_t stream) {
    const int*   node_feat = (const int*)  d_in[0];
    const int*   edge_feat = (const int*)  d_in[1];
    const int*   edge_index= (const int*)  d_in[2];
    const float* node_emb  = (const float*)d_in[3];
    const float* edge_emb  = (const float*)d_in[4];
    const float* W_t       = (const float*)d_in[5];
    const float* b_t       = (const float*)d_in[6];
    const float* W_e       = (const float*)d_in[7];
    const float* b_e       = (const float*)d_in[8];
    const float* a_w       = (const float*)d_in[9];
    const float* a_b       = (const float*)d_in[10];
    const float* ln_gamma  = (const float*)d_in[11];
    const float* ln_beta   = (const float*)d_in[12];

    char* ws = (char*)d_ws;
    float*    proj_node = (float*)   (ws + OFF_PROJ_NODE);
    float*    proj_edge = (float*)   (ws + OFF_PROJ_EDGE);
    float*    ln_edge   = (float*)   (ws + OFF_LN_EDGE);
    float*    att1      = (float*)   (ws + OFF_ATT1);
    float*    att2      = (float*)   (ws + OFF_ATT2);
    float*    att3      = (float*)   (ws + OFF_ATT3);
    unsigned* segmax    = (unsigned*)(ws + OFF_SEGMAX);
    float*    segsum    = (float*)   (ws + OFF_SEGSUM);
    int*      indeg     = (int*)     (ws + OFF_INDEG);
    float*    eij       = (float*)   (ws + OFF_EIJ);
    float*    weight    = (float*)   (ws + OFF_WEIGHT);

    float* out_node = (float*)d_out;                         // 25000 x 128
    float* edge_out = (float*)d_out + (size_t)N_NODES * HF;  // 400000 x 128

    const bool factored = (ws_size >= WS_FACTORED);

    // init: segmax (0 == ordered -inf), segsum, indeg  (contiguous region)
    hipMemsetAsync(ws + OFF_SEGMAX, 0, OFF_EIJ - OFF_SEGMAX, stream);
    if (factored)
        hipMemsetAsync(weight, 0, (size_t)HEADS * N_NODES * NDICT_PAD * 4, stream);
    else
        hipMemsetAsync(out_node, 0, (size_t)N_NODES * HF * 4, stream);

    // dictionary GEMMs (WMMA f32 16x16x4)
    k_wmma_dict_gemm<<<dim3(HF / 16, NDICT_PAD / 16), 32, 0, stream>>>(
        node_emb, N_NDICT, W_t, b_t, proj_node);
    k_wmma_dict_gemm<<<dim3(HF / 16, 1), 32, 0, stream>>>(
        edge_emb, N_EDICT, W_e, b_e, proj_edge);

    // attention tables + edge LN table
    k_att_node<<<N_NDICT, HF, 0, stream>>>(proj_node, a_w, att1, att2);
    k_edge_tables<<<N_EDICT, HF, 0, stream>>>(proj_edge, a_w, ln_gamma, ln_beta,
                                              att3, ln_edge);

    // per-edge logits + segment max + indegree
    k_logits<<<(N_EDGES + 255) / 256, 256, 0, stream>>>(
        edge_index, edge_feat, node_feat, att1, att2, att3, a_b,
        eij, segmax, indeg);

    // edge output stream (pure gather of 16-row LN table)
    k_edge_out<<<(N_EDGES * (HF / 4) + 255) / 256, 256, 0, stream>>>(
        edge_feat, (const float4*)ln_edge, (float4*)edge_out);

    // softmax denominator
    k_expsum<<<(N_EDGES * HEADS + 255) / 256, 256, 0, stream>>>(
        edge_index, eij, segmax, segsum);

    if (factored) {
        // scatter alpha into dict-factored weights (1.6M atomics, L2-resident)
        k_alpha_weights<<<(N_EDGES * HEADS + 255) / 256, 256, 0, stream>>>(
            edge_index, node_feat, eij, segsum, weight);
        // dense recovery GEMM: per head 25000x112 @ 112x32  (WMMA f32)
        k_wmma_recover<<<dim3(HEADS, (N_NODES + 15) / 16), 32, 0, stream>>>(
            weight, proj_node, out_node);
    } else {
        // fallback: direct message scatter (51M atomics)
        k_msg_direct<<<(N_EDGES * HF + 255) / 256, 256, 0, stream>>>(
            edge_index, node_feat, eij, segsum, proj_node, out_node);
    }

    // + indeg * h_i, then LayerNorm
    k_finalize_node<<<N_NODES, HF, 0, stream>>>(
        node_feat, indeg, proj_node, ln_gamma, ln_beta, out_node);
}